// HAN_50362786513106
// MI455X (gfx1250) — compile-verified
//
#include <hip/hip_runtime.h>
#include <float.h>
#include <math.h>

// ---------------------------------------------------------------------------
// HANConv on MI455X (gfx1250). fp32 WMMA (v_wmma_f32_16x16x4_f32) for all
// GEMM stages; edge aggregation via L2-resident float atomics.
// ---------------------------------------------------------------------------

typedef __attribute__((ext_vector_type(2))) float v2f;
typedef __attribute__((ext_vector_type(8))) float v8f;

#define NMAT 100000
#define NELEM 118
#define HIDC 128

__device__ __forceinline__ v8f wmma4(v2f a, v2f b, v8f c) {
  // D(16x16,f32) = A(16x4,f32) x B(4x16,f32) + C
  return __builtin_amdgcn_wmma_f32_16x16x4_f32(
      /*neg_a=*/false, a, /*neg_b=*/false, b,
      /*c_mod=*/(short)0, c, /*reuse_a=*/false, /*reuse_b=*/false);
}

__device__ __forceinline__ float leaky02(float x) { return x >= 0.f ? x : 0.2f * x; }

__device__ __forceinline__ void atomicMaxF(float* addr, float v) {
  if (v >= 0.f) (void)atomicMax((int*)addr, __float_as_int(v));
  else          (void)atomicMin((unsigned int*)addr, __float_as_uint(v));
}

// ---------------------------------------------------------------------------
__global__ void fill_kernel(float* __restrict__ p, float v, size_t n) {
  size_t i = (size_t)blockIdx.x * blockDim.x + threadIdx.x;
  if (i < n) p[i] = v;
}

// C[M x N] = A[M x K] * B[N x K]^T + bias[N]
// grid.x = ceil(M/16); blockDim.x = 32 * (N/16); one 16x16 tile per wave.
__global__ void gemm_bias_kernel(const float* __restrict__ A, const float* __restrict__ B,
                                 const float* __restrict__ bias, float* __restrict__ C,
                                 int M, int K, int N) {
  const int lane = threadIdx.x & 31;
  const int wt   = threadIdx.x >> 5;   // column tile
  const int mt   = blockIdx.x;         // row tile
  const int r16  = lane & 15;
  const int kh   = lane >> 4;          // K half-select (ISA A-frag layout)
  int arow = mt * 16 + r16;
  if (arow > M - 1) arow = M - 1;      // clamp, keep EXEC all-ones for WMMA
  const float* Ap = A + (size_t)arow * K + 2 * kh;
  const float* Bp = B + (size_t)(wt * 16 + r16) * K + 2 * kh;
  v8f acc = {};
  for (int k0 = 0; k0 < K; k0 += 4) {
    v2f a = *(const v2f*)(Ap + k0);
    v2f b = *(const v2f*)(Bp + k0);
    acc = wmma4(a, b, acc);
  }
  const float bv = bias[wt * 16 + r16];
  const int cg = wt * 16 + r16;
#pragma unroll
  for (int v = 0; v < 8; ++v) {
    int r = mt * 16 + 8 * kh + v;      // C/D layout: row = vgpr + 8*(lane>=16)
    if (r < M) C[(size_t)r * N + cg] = acc[v] + bv;
  }
}

// colsum[wt*16+col] += sum_over_tile_rows tanh( relu(A) @ B^T + bias )
// M must be a multiple of 16. K = N = 128. blockDim.x = 256.
__global__ void tanh_colsum_kernel(const float* __restrict__ A, const float* __restrict__ B,
                                   const float* __restrict__ bias, float* __restrict__ colsum,
                                   int K) {
  const int lane = threadIdx.x & 31;
  const int wt   = threadIdx.x >> 5;
  const int mt   = blockIdx.x;
  const int r16  = lane & 15;
  const int kh   = lane >> 4;
  const float* Ap = A + (size_t)(mt * 16 + r16) * K + 2 * kh;
  const float* Bp = B + (size_t)(wt * 16 + r16) * K + 2 * kh;
  v8f acc = {};
  for (int k0 = 0; k0 < K; k0 += 4) {
    v2f a = *(const v2f*)(Ap + k0);
    a[0] = fmaxf(a[0], 0.f);           // fused ReLU from edge-conv epilogue
    a[1] = fmaxf(a[1], 0.f);
    v2f b = *(const v2f*)(Bp + k0);
    acc = wmma4(a, b, acc);
  }
  const float bv = bias[wt * 16 + r16];
  float p = 0.f;
#pragma unroll
  for (int v = 0; v < 8; ++v) p += tanhf(acc[v] + bv);
  p += __shfl_xor(p, 16, 32);          // combine row-halves (wave32)
  if (lane < 16) atomicAdd(&colsum[wt * 16 + lane], p);
}

// out[M x N] = (attn0*relu(o0) + attn1*relu(o1)) @ Wl^T + bl.  N=64 -> 128 thr.
__global__ void final_gemm_kernel(const float* __restrict__ o0, const float* __restrict__ o1,
                                  const float* __restrict__ attn,
                                  const float* __restrict__ Wl, const float* __restrict__ bl,
                                  float* __restrict__ out, int K, int N) {
  const float w0 = attn[0], w1 = attn[1];
  const int lane = threadIdx.x & 31;
  const int wt   = threadIdx.x >> 5;
  const int mt   = blockIdx.x;
  const int r16  = lane & 15;
  const int kh   = lane >> 4;
  const size_t abase = (size_t)(mt * 16 + r16) * K + 2 * kh;
  const float* Bp = Wl + (size_t)(wt * 16 + r16) * K + 2 * kh;
  v8f acc = {};
  for (int k0 = 0; k0 < K; k0 += 4) {
    v2f a0 = *(const v2f*)(o0 + abase + k0);
    v2f a1 = *(const v2f*)(o1 + abase + k0);
    v2f a;
    a[0] = w0 * fmaxf(a0[0], 0.f) + w1 * fmaxf(a1[0], 0.f);
    a[1] = w0 * fmaxf(a0[1], 0.f) + w1 * fmaxf(a1[1], 0.f);
    v2f b = *(const v2f*)(Bp + k0);
    acc = wmma4(a, b, acc);
  }
  const float bv = bl[wt * 16 + r16];
  const int cg = wt * 16 + r16;
#pragma unroll
  for (int v = 0; v < 8; ++v)
    out[(size_t)(mt * 16 + 8 * kh + v) * N + cg] = acc[v] + bv;
}

// per-node per-head attention dot:  out[n*8+h] = sum_d h[n,h,d] * a[h,d]
__global__ void node_alpha_kernel(const float* __restrict__ h, const float* __restrict__ av,
                                  float* __restrict__ out, int N) {
  int t = blockIdx.x * blockDim.x + threadIdx.x;
  if (t >= N * 8) return;
  int n = t >> 3, hd = t & 7;
  const float* hp = h + (size_t)n * HIDC + hd * 16;
  const float* ap = av + hd * 16;
  float s = 0.f;
#pragma unroll
  for (int d = 0; d < 16; ++d) s += hp[d] * ap[d];
  out[t] = s;
}

__global__ void edge_max_kernel(const int* __restrict__ src, const int* __restrict__ dst,
                                const float* __restrict__ as, const float* __restrict__ ad,
                                float* __restrict__ amax, int EH) {
  int t = blockIdx.x * blockDim.x + threadIdx.x;
  if (t >= EH) return;
  int e = t >> 3, hh = t & 7;
  int s = src[e], d = dst[e];
  float a = leaky02(as[s * 8 + hh] + ad[d * 8 + hh]);
  atomicMaxF(&amax[d * 8 + hh], a);
}

__global__ void edge_den_kernel(const int* __restrict__ src, const int* __restrict__ dst,
                                const float* __restrict__ as, const float* __restrict__ ad,
                                const float* __restrict__ amax, float* __restrict__ den, int EH) {
  int t = blockIdx.x * blockDim.x + threadIdx.x;
  if (t >= EH) return;
  int e = t >> 3, hh = t & 7;
  int s = src[e], d = dst[e];
  float a = leaky02(as[s * 8 + hh] + ad[d * 8 + hh]);
  atomicAdd(&den[d * 8 + hh], expf(a - amax[d * 8 + hh]));
}

// one 128-thread block per edge: out[dst,:] += x_s[src,:] * w[head]
__global__ void edge_scatter_kernel(const int* __restrict__ src, const int* __restrict__ dst,
                                    const float* __restrict__ as, const float* __restrict__ ad,
                                    const float* __restrict__ amax, const float* __restrict__ den,
                                    const float* __restrict__ xs, float* __restrict__ out) {
  int e = blockIdx.x;
  int c = threadIdx.x;                 // 0..127
  int hh = c >> 4;
  int s = src[e], d = dst[e];
  float a = leaky02(as[s * 8 + hh] + ad[d * 8 + hh]);
  float w = expf(a - amax[d * 8 + hh]) / (den[d * 8 + hh] + 1e-16f);
  atomicAdd(&out[(size_t)d * HIDC + c], xs[(size_t)s * HIDC + c] * w);
}

// semantic softmax over 2 metapaths: score[m] = (q . colsum[m]) / N
__global__ void scores_kernel(const float* __restrict__ colsum, const float* __restrict__ q,
                              float* __restrict__ attn, float invN) {
  int lane = threadIdx.x;              // 32 threads
  float s0 = 0.f, s1 = 0.f;
  for (int c = lane; c < HIDC; c += 32) {
    s0 += q[c] * colsum[c];
    s1 += q[c] * colsum[HIDC + c];
  }
  for (int o = 16; o > 0; o >>= 1) {
    s0 += __shfl_xor(s0, o, 32);
    s1 += __shfl_xor(s1, o, 32);
  }
  if (lane == 0) {
    s0 *= invN; s1 *= invN;
    float m = fmaxf(s0, s1);
    float e0 = expf(s0 - m), e1 = expf(s1 - m);
    float dsum = e0 + e1;
    attn[0] = e0 / dsum;
    attn[1] = e1 / dsum;
  }
}

// ---------------------------------------------------------------------------
extern "C" void kernel_launch(void* const* d_in, const int* in_sizes, int n_in,
                              void* d_out, int out_size, void* d_ws, size_t ws_size,
                              hipStream_t stream) {
  (void)n_in; (void)out_size; (void)ws_size;
  const float* x_mat       = (const float*)d_in[0];
  const float* x_elem      = (const float*)d_in[1];
  const float* W_proj_mat  = (const float*)d_in[2];
  const float* b_proj_mat  = (const float*)d_in[3];
  const float* W_proj_elem = (const float*)d_in[4];
  const float* b_proj_elem = (const float*)d_in[5];
  const float* a_src_em    = (const float*)d_in[6];
  const float* a_dst_em    = (const float*)d_in[7];
  const float* a_src_mm    = (const float*)d_in[8];
  const float* a_dst_mm    = (const float*)d_in[9];
  const float* Wk          = (const float*)d_in[10];
  const float* bk          = (const float*)d_in[11];
  const float* q           = (const float*)d_in[12];
  const float* Wl          = (const float*)d_in[13];
  const float* bl          = (const float*)d_in[14];
  const int*   src_em      = (const int*)d_in[15];
  const int*   dst_em      = (const int*)d_in[16];
  const int*   src_mm      = (const int*)d_in[17];
  const int*   dst_mm      = (const int*)d_in[18];
  const int E_em = in_sizes[15];
  const int E_mm = in_sizes[17];
  float* out = (float*)d_out;

  // ---- workspace layout (floats) ----
  float* ws = (float*)d_ws;
  size_t off = 0;
  float* h_mat   = ws + off; off += (size_t)NMAT  * HIDC;
  float* h_elem  = ws + off; off += (size_t)NELEM * HIDC;
  float* as_em   = ws + off; off += (size_t)NELEM * 8;
  float* ad_em   = ws + off; off += (size_t)NMAT  * 8;
  float* as_mm   = ws + off; off += (size_t)NMAT  * 8;
  float* ad_mm   = ws + off; off += (size_t)NMAT  * 8;
  float* amax_em = ws + off; off += (size_t)NMAT  * 8;   // } contiguous: one -FLT_MAX fill
  float* amax_mm = ws + off; off += (size_t)NMAT  * 8;   // }
  float* den_em  = ws + off; off += (size_t)NMAT  * 8;   // } contiguous: one zero fill
  float* den_mm  = ws + off; off += (size_t)NMAT  * 8;   // }
  float* o_em    = ws + off; off += (size_t)NMAT  * HIDC;// }
  float* o_mm    = ws + off; off += (size_t)NMAT  * HIDC;// }
  float* colsum  = ws + off; off += 256;                 // }
  float* attn    = ws + off; off += 2;                   // }

  // ---- init ----
  {
    size_t nmax = (size_t)NMAT * 8 * 2;
    fill_kernel<<<(unsigned)((nmax + 255) / 256), 256, 0, stream>>>(amax_em, -FLT_MAX, nmax);
    size_t nz = (size_t)NMAT * 8 * 2 + (size_t)NMAT * HIDC * 2 + 258;
    fill_kernel<<<(unsigned)((nz + 255) / 256), 256, 0, stream>>>(den_em, 0.f, nz);
  }

  // ---- projections (fp32 WMMA GEMM) ----
  gemm_bias_kernel<<<NMAT / 16, 256, 0, stream>>>(x_mat, W_proj_mat, b_proj_mat, h_mat,
                                                  NMAT, 128, HIDC);
  gemm_bias_kernel<<<(NELEM + 15) / 16, 256, 0, stream>>>(x_elem, W_proj_elem, b_proj_elem,
                                                          h_elem, NELEM, 64, HIDC);

  // ---- per-node attention dots ----
  node_alpha_kernel<<<(NELEM * 8 + 255) / 256, 256, 0, stream>>>(h_elem, a_src_em, as_em, NELEM);
  node_alpha_kernel<<<(NMAT * 8 + 255) / 256, 256, 0, stream>>>(h_mat, a_dst_em, ad_em, NMAT);
  node_alpha_kernel<<<(NMAT * 8 + 255) / 256, 256, 0, stream>>>(h_mat, a_src_mm, as_mm, NMAT);
  node_alpha_kernel<<<(NMAT * 8 + 255) / 256, 256, 0, stream>>>(h_mat, a_dst_mm, ad_mm, NMAT);

  // ---- elem->mat edge conv ----
  {
    int EH = E_em * 8;
    edge_max_kernel<<<(EH + 255) / 256, 256, 0, stream>>>(src_em, dst_em, as_em, ad_em, amax_em, EH);
    edge_den_kernel<<<(EH + 255) / 256, 256, 0, stream>>>(src_em, dst_em, as_em, ad_em, amax_em, den_em, EH);
    edge_scatter_kernel<<<E_em, 128, 0, stream>>>(src_em, dst_em, as_em, ad_em, amax_em, den_em,
                                                  h_elem, o_em);
  }
  // ---- mat->mat edge conv ----
  {
    int EH = E_mm * 8;
    edge_max_kernel<<<(EH + 255) / 256, 256, 0, stream>>>(src_mm, dst_mm, as_mm, ad_mm, amax_mm, EH);
    edge_den_kernel<<<(EH + 255) / 256, 256, 0, stream>>>(src_mm, dst_mm, as_mm, ad_mm, amax_mm, den_mm, EH);
    edge_scatter_kernel<<<E_mm, 128, 0, stream>>>(src_mm, dst_mm, as_mm, ad_mm, amax_mm, den_mm,
                                                  h_mat, o_mm);
  }

  // ---- semantic attention (WMMA GEMM + tanh/colsum epilogue) ----
  tanh_colsum_kernel<<<NMAT / 16, 256, 0, stream>>>(o_em, Wk, bk, colsum, 128);
  tanh_colsum_kernel<<<NMAT / 16, 256, 0, stream>>>(o_mm, Wk, bk, colsum + HIDC, 128);
  scores_kernel<<<1, 32, 0, stream>>>(colsum, q, attn, 1.f / (float)NMAT);

  // ---- final combine + Linear (WMMA GEMM) ----
  final_gemm_kernel<<<NMAT / 16, 128, 0, stream>>>(o_em, o_mm, attn, Wl, bl, out, 128, 64);
}